// GAT_79070347919663
// MI455X (gfx1250) — compile-verified
//
#include <hip/hip_runtime.h>
#include <math.h>

// Problem constants (from reference)
#define F_IN 256
#define HEADS1 4
#define CH1 32
#define HC1 128   // HEADS1*CH1
#define CLS 16

typedef __attribute__((ext_vector_type(2))) float v2f;
typedef __attribute__((ext_vector_type(8))) float v8f;

// ---------------------------------------------------------------------------
// Utility: fill buffer with a constant
// ---------------------------------------------------------------------------
__global__ void fill_f32(float* __restrict__ p, float v, size_t n) {
  size_t i = (size_t)blockIdx.x * blockDim.x + threadIdx.x;
  if (i < n) p[i] = v;
}

// ---------------------------------------------------------------------------
// FP32 WMMA GEMM: C[M,Ncols] = A[M,K] @ B[K,Ncols], all row-major.
// One wave32 computes a 16 x (16*NT) strip: the A fragment is loaded once per
// K-step and fed to NT independent V_WMMA_F32_16X16X4_F32 accumulation chains
// (A-gather traffic /NT, and NT-way ILP across the XDL pipe instead of a
// single serial C-dependency chain).
// Fragment layouts per CDNA5 ISA 7.12.2:
//   A 16x4:  lanes 0-15 row M=lane, VGPR0/1 = K0/K1; lanes 16-31 same rows, K2/K3
//   B 4x16:  lanes 0-15 col N=lane, rows K0/K1; lanes 16-31 rows K2/K3
//   C 16x16: VGPR r -> row r (lanes 0-15) / row r+8 (lanes 16-31), col = lane&15
// ---------------------------------------------------------------------------
template <int NT>
__global__ __launch_bounds__(32) void wmma_gemm_f32_nt(
    const float* __restrict__ A, const float* __restrict__ B,
    float* __restrict__ C, int M, int K, int Ncols) {
  const int lane   = threadIdx.x;
  const int tileN0 = blockIdx.x * NT;              // first 16-col tile of strip
  const int tileM  = blockIdx.y;

  const int mrow = tileM * 16 + (lane & 15);
  const int mr   = mrow < M ? mrow : (M - 1);      // clamp (M%16==0 here anyway)
  const int ksub = (lane >> 4) << 1;               // 0 or 2

  const float* __restrict__ Arow = A + (size_t)mr * K;
  const float* __restrict__ Bp   = B + (size_t)tileN0 * 16 + (lane & 15);

  v8f acc[NT];
#pragma unroll
  for (int t = 0; t < NT; ++t) acc[t] = (v8f){};

  for (int k0 = 0; k0 < K; k0 += 4) {
    v2f a;
    a.x = Arow[k0 + ksub];
    a.y = Arow[k0 + ksub + 1];
    const float* __restrict__ Brow0 = Bp + (size_t)(k0 + ksub) * Ncols;
    const float* __restrict__ Brow1 = Bp + (size_t)(k0 + ksub + 1) * Ncols;
#pragma unroll
    for (int t = 0; t < NT; ++t) {
      v2f b;
      b.x = Brow0[t * 16];
      b.y = Brow1[t * 16];
      acc[t] = __builtin_amdgcn_wmma_f32_16x16x4_f32(
          /*neg_a=*/false, a, /*neg_b=*/false, b,
          /*c_mod=*/(short)0, acc[t], /*reuse_a=*/false, /*reuse_b=*/false);
    }
  }

  const int rbase = tileM * 16 + ((lane >> 4) << 3);
  const int colb  = (lane & 15);
#pragma unroll
  for (int t = 0; t < NT; ++t) {
    const int col = (tileN0 + t) * 16 + colb;
#pragma unroll
    for (int r = 0; r < 8; ++r) {
      int ro = rbase + r;
      if (ro < M) C[(size_t)ro * Ncols + col] = acc[t][r];
    }
  }
}

// ---------------------------------------------------------------------------
// Per-(node,head) attention logits: alpha = <h[n,head,:], att[head,:]>
// ---------------------------------------------------------------------------
template <int HEADS, int CH>
__global__ void alpha_kernel(const float* __restrict__ h,
                             const float* __restrict__ att_s,
                             const float* __restrict__ att_d,
                             float* __restrict__ as, float* __restrict__ ad,
                             int n) {
  int i = blockIdx.x * blockDim.x + threadIdx.x;
  if (i >= n * HEADS) return;
  int node = i / HEADS, hd = i % HEADS;
  const float* hp = h + (size_t)node * (HEADS * CH) + (size_t)hd * CH;
  const float* sp = att_s + hd * CH;
  const float* dp = att_d + hd * CH;
  float s = 0.f, d = 0.f;
#pragma unroll
  for (int c = 0; c < CH; ++c) {
    float v = hp[c];
    s += v * sp[c];
    d += v * dp[c];
  }
  as[i] = s;
  ad[i] = d;
}

// ---------------------------------------------------------------------------
// Edge helpers
// ---------------------------------------------------------------------------
__device__ inline void edge_sd(int i, const int* __restrict__ ei, int E,
                               int& s, int& d) {
  if (i < E) { s = ei[i]; d = ei[E + i]; }
  else       { s = i - E; d = s; }          // appended self-loops
}

__device__ inline float lrelu(float e) { return e > 0.f ? e : 0.2f * e; }

// float atomic max via signed/unsigned int monotonicity (buffers init to -inf)
__device__ inline void atomicMaxFloat(float* addr, float val) {
  if (val >= 0.f) atomicMax((int*)addr, __float_as_int(val));
  else            atomicMin((unsigned int*)addr, __float_as_uint(val));
}

// Pass 1: segment max of leaky_relu(alpha_s[src]+alpha_d[dst]) over dst
template <int HEADS>
__global__ void edge_max(const int* __restrict__ ei, int E, int nE,
                         const float* __restrict__ as, const float* __restrict__ ad,
                         float* __restrict__ m) {
  int i = blockIdx.x * blockDim.x + threadIdx.x;
  if (i >= nE) return;
  int s, d; edge_sd(i, ei, E, s, d);
#pragma unroll
  for (int h = 0; h < HEADS; ++h) {
    float e = lrelu(as[s * HEADS + h] + ad[d * HEADS + h]);
    atomicMaxFloat(&m[d * HEADS + h], e);
  }
}

// Pass 2: segment sum of exp(e - m[dst])
template <int HEADS>
__global__ void edge_den(const int* __restrict__ ei, int E, int nE,
                         const float* __restrict__ as, const float* __restrict__ ad,
                         const float* __restrict__ m, float* __restrict__ den) {
  int i = blockIdx.x * blockDim.x + threadIdx.x;
  if (i >= nE) return;
  int s, d; edge_sd(i, ei, E, s, d);
#pragma unroll
  for (int h = 0; h < HEADS; ++h) {
    float e  = lrelu(as[s * HEADS + h] + ad[d * HEADS + h]);
    float ex = expf(e - m[d * HEADS + h]);
    atomicAdd(&den[d * HEADS + h], ex);
  }
}

// Pass 3: agg[dst] += h[src] * coef.  (HEADS*CH)/4 lanes per edge, float4 gather.
template <int HEADS, int CH>
__global__ void edge_agg(const int* __restrict__ ei, int E, int nE,
                         const float* __restrict__ as, const float* __restrict__ ad,
                         const float* __restrict__ m, const float* __restrict__ den,
                         const float* __restrict__ h, float* __restrict__ agg) {
  constexpr int TC  = HEADS * CH;   // total channels
  constexpr int LPE = TC >> 2;      // lanes per edge (4 channels per lane)
  long long t = (long long)blockIdx.x * blockDim.x + threadIdx.x;
  long long i = t / LPE;            // compile-time power-of-two -> shift
  if (i >= nE) return;
  int lane = (int)(t & (LPE - 1));
  int c0   = lane << 2;
  int hd   = c0 / CH;
  int s, d; edge_sd((int)i, ei, E, s, d);
  float e    = lrelu(as[s * HEADS + hd] + ad[d * HEADS + hd]);
  float coef = expf(e - m[d * HEADS + hd]) / (den[d * HEADS + hd] + 1e-16f);
  const float4 hv = *(const float4*)(h + (size_t)s * TC + c0);
  float* ap = agg + (size_t)d * TC + c0;
  atomicAdd(ap + 0, hv.x * coef);
  atomicAdd(ap + 1, hv.y * coef);
  atomicAdd(ap + 2, hv.z * coef);
  atomicAdd(ap + 3, hv.w * coef);
}

// h2 = elu(agg + b)
__global__ void elu_bias(const float* __restrict__ agg, const float* __restrict__ b,
                         float* __restrict__ out, size_t total, int tc) {
  size_t i = (size_t)blockIdx.x * blockDim.x + threadIdx.x;
  if (i >= total) return;
  float v = agg[i] + b[i % tc];
  out[i] = v > 0.f ? v : (expf(v) - 1.f);
}

// out = agg + b
__global__ void add_bias(const float* __restrict__ agg, const float* __restrict__ b,
                         float* __restrict__ out, size_t total, int tc) {
  size_t i = (size_t)blockIdx.x * blockDim.x + threadIdx.x;
  if (i >= total) return;
  out[i] = agg[i] + b[i % tc];
}

// ---------------------------------------------------------------------------
// Launch
// ---------------------------------------------------------------------------
static inline unsigned cdiv(size_t a, size_t b) { return (unsigned)((a + b - 1) / b); }

extern "C" void kernel_launch(void* const* d_in, const int* in_sizes, int n_in,
                              void* d_out, int out_size, void* d_ws, size_t ws_size,
                              hipStream_t stream) {
  const float* x   = (const float*)d_in[0];
  const int*   ei  = (const int*)d_in[1];
  const float* W1  = (const float*)d_in[2];
  const float* as1 = (const float*)d_in[3];
  const float* ad1 = (const float*)d_in[4];
  const float* b1  = (const float*)d_in[5];
  const float* W2  = (const float*)d_in[6];
  const float* as2 = (const float*)d_in[7];
  const float* ad2 = (const float*)d_in[8];
  const float* b2  = (const float*)d_in[9];
  float* out = (float*)d_out;

  const int N  = in_sizes[0] / F_IN;   // 50000
  const int E  = in_sizes[1] / 2;      // 800000
  const int nE = E + N;                // with self-loops

  // Workspace carve-up (floats). h2 aliases h1 (h1 dead after edge_agg);
  // g2 aliases agg1 (agg1 dead after elu_bias).
  float* ws = (float*)d_ws;
  size_t off = 0;
  float* h1   = ws + off; off += (size_t)N * HC1;   // layer-1 features (then h2)
  float* aS1  = ws + off; off += (size_t)N * HEADS1;
  float* aD1  = ws + off; off += (size_t)N * HEADS1;
  float* m1   = ws + off; off += (size_t)N * HEADS1;
  float* dn1  = ws + off; off += (size_t)N * HEADS1;
  float* agg1 = ws + off; off += (size_t)N * HC1;   // layer-1 aggregate (then g2)
  float* aS2  = ws + off; off += (size_t)N;
  float* aD2  = ws + off; off += (size_t)N;
  float* m2   = ws + off; off += (size_t)N;
  float* dn2  = ws + off; off += (size_t)N;
  float* agg2 = ws + off; off += (size_t)N * CLS;
  float* h2 = h1;     // reuse
  float* g2 = agg1;   // reuse

  const int B = 256;
  const float NEG_INF = -INFINITY;

  // Init accumulators
  fill_f32<<<cdiv((size_t)N * HEADS1, B), B, 0, stream>>>(m1,  NEG_INF, (size_t)N * HEADS1);
  fill_f32<<<cdiv((size_t)N * HEADS1, B), B, 0, stream>>>(dn1, 0.f,     (size_t)N * HEADS1);
  fill_f32<<<cdiv((size_t)N * HC1,   B), B, 0, stream>>>(agg1, 0.f,     (size_t)N * HC1);

  // ---- Layer 1 ----
  {
    dim3 grid(HC1 / (16 * 8), cdiv((size_t)N, 16));     // 16x128 strip per wave
    wmma_gemm_f32_nt<8><<<grid, 32, 0, stream>>>(x, W1, h1, N, F_IN, HC1);
  }
  alpha_kernel<HEADS1, CH1><<<cdiv((size_t)N * HEADS1, B), B, 0, stream>>>(h1, as1, ad1, aS1, aD1, N);
  edge_max<HEADS1><<<cdiv((size_t)nE, B), B, 0, stream>>>(ei, E, nE, aS1, aD1, m1);
  edge_den<HEADS1><<<cdiv((size_t)nE, B), B, 0, stream>>>(ei, E, nE, aS1, aD1, m1, dn1);
  edge_agg<HEADS1, CH1><<<cdiv((size_t)nE * (HC1 / 4), B), B, 0, stream>>>(ei, E, nE, aS1, aD1,
                                                                           m1, dn1, h1, agg1);
  elu_bias<<<cdiv((size_t)N * HC1, B), B, 0, stream>>>(agg1, b1, h2, (size_t)N * HC1, HC1);

  // ---- Layer 2 ----
  fill_f32<<<cdiv((size_t)N, B), B, 0, stream>>>(m2,  NEG_INF, (size_t)N);
  fill_f32<<<cdiv((size_t)N, B), B, 0, stream>>>(dn2, 0.f,     (size_t)N);
  fill_f32<<<cdiv((size_t)N * CLS, B), B, 0, stream>>>(agg2, 0.f, (size_t)N * CLS);
  {
    dim3 grid(CLS / 16, cdiv((size_t)N, 16));           // single 16-col tile
    wmma_gemm_f32_nt<1><<<grid, 32, 0, stream>>>(h2, W2, g2, N, HC1, CLS);
  }
  alpha_kernel<1, CLS><<<cdiv((size_t)N, B), B, 0, stream>>>(g2, as2, ad2, aS2, aD2, N);
  edge_max<1><<<cdiv((size_t)nE, B), B, 0, stream>>>(ei, E, nE, aS2, aD2, m2);
  edge_den<1><<<cdiv((size_t)nE, B), B, 0, stream>>>(ei, E, nE, aS2, aD2, m2, dn2);
  edge_agg<1, CLS><<<cdiv((size_t)nE * (CLS / 4), B), B, 0, stream>>>(ei, E, nE, aS2, aD2,
                                                                      m2, dn2, g2, agg2);
  add_bias<<<cdiv((size_t)N * CLS, B), B, 0, stream>>>(agg2, b2, out, (size_t)N * CLS, CLS);
}